// GraspNet_33612414058695
// MI455X (gfx1250) — compile-verified
//
#include <hip/hip_runtime.h>
#include <hip/hip_bf16.h>

#define B_ 4
#define N_ 20000
#define C_ 512
#define M_ 1024
#define GRASP_TH 0.1f

typedef float v2f __attribute__((ext_vector_type(2)));
typedef float v8f __attribute__((ext_vector_type(8)));

#if defined(__gfx1250__) && __has_builtin(__builtin_amdgcn_wmma_f32_16x16x4_f32)
#define USE_WMMA 1
#else
#define USE_WMMA 0
#endif

// ---------------------------------------------------------------------------
// Branch-free sorted insert into ascending top-3, STREAMING variant: caller
// guarantees candidate indices arrive in strictly ascending order, so plain
// strict float '<' implements the lax.top_k tie-break (lower index wins) for
// free. Bitwise bool ops (no ||/&&) -> v_cmp + v_cndmask, no EXEC writes.
// ---------------------------------------------------------------------------
__device__ __forceinline__ void ins3_stream(float v, int n, float d[3], int ix[3]) {
  const bool c0 = v < d[0];
  const bool c1 = v < d[1];
  const bool c2 = v < d[2];
  const float r0 = c0 ? v     : d[0];
  const int   j0 = c0 ? n     : ix[0];
  const float r1 = c0 ? d[0]  : (c1 ? v : d[1]);
  const int   j1 = c0 ? ix[0] : (c1 ? n : ix[1]);
  const float r2 = c1 ? d[1]  : (c2 ? v : d[2]);
  const int   j2 = c1 ? ix[1] : (c2 ? n : ix[2]);
  d[0] = r0; d[1] = r1; d[2] = r2;
  ix[0] = j0; ix[1] = j1; ix[2] = j2;
}

// Full lexicographic variant for cross-lane merges (indices unordered there).
// Bitwise | and & avoid short-circuit EXEC manipulation.
__device__ __forceinline__ void ins3_lex(float v, int n, float d[3], int ix[3]) {
  const bool c0 = (v < d[0]) | ((v == d[0]) & (n < ix[0]));
  const bool c1 = (v < d[1]) | ((v == d[1]) & (n < ix[1]));
  const bool c2 = (v < d[2]) | ((v == d[2]) & (n < ix[2]));
  const float r0 = c0 ? v     : d[0];
  const int   j0 = c0 ? n     : ix[0];
  const float r1 = c0 ? d[0]  : (c1 ? v : d[1]);
  const int   j1 = c0 ? ix[0] : (c1 ? n : ix[1]);
  const float r2 = c1 ? d[1]  : (c2 ? v : d[2]);
  const int   j2 = c1 ? ix[1] : (c2 ? n : ix[2]);
  d[0] = r0; d[1] = r1; d[2] = r2;
  ix[0] = j0; ix[1] = j1; ix[2] = j2;
}

// ---------------------------------------------------------------------------
// Kernel 1: masked furthest-point sampling, one 32-wave workgroup per batch.
// CDNA5 WGP LDS (320 KB) holds the WHOLE per-batch working set:
//   s_xyz  240,000 B  (point cloud, loaded once in phase 0)
//   s_mind  80,000 B  (running min distance; mask folded into the init value:
//                      unmasked points start at -inf, so fminf keeps them at
//                      -inf forever and no separate mask array is needed)
// The 1024 serial steps then touch no global memory except writing one index.
// argmax==1 <=> o1 > o0 strictly (jnp.argmax ties resolve to index 0).
// ---------------------------------------------------------------------------
__global__ __launch_bounds__(1024) void fps_kernel(
    const float* __restrict__ pc, const float* __restrict__ obj,
    const float* __restrict__ grasp, int* __restrict__ fps_idx) {
  const int b    = blockIdx.x;
  const int tid  = threadIdx.x;
  const int lane = tid & 31;
  const int wid  = tid >> 5;

  __shared__ float s_xyz[N_ * 3];   // 240000 B
  __shared__ float s_mind[N_];      //  80000 B
  __shared__ float swv[32];
  __shared__ int   swi[32];
  __shared__ int   s_last;
  __shared__ float s_px, s_py, s_pz;

  const float* pcb = pc + (size_t)b * N_ * 3;
  const float* ob0 = obj + (size_t)b * 2 * N_;
  const float* gr  = grasp + (size_t)b * N_;

  // ---- phase 0: stage xyz to LDS, init mind(+mask), find first masked ----
  int firstLoc = N_;
  for (int n = tid; n < N_; n += 1024) {
    const float* p = pcb + (size_t)n * 3;
    const float x = p[0], y = p[1], z = p[2];
    s_xyz[n * 3 + 0] = x;
    s_xyz[n * 3 + 1] = y;
    s_xyz[n * 3 + 2] = z;
    const bool mk = (ob0[N_ + n] > ob0[n]) & (gr[n] > GRASP_TH);
    s_mind[n] = mk ? __builtin_inff() : -__builtin_inff();
    firstLoc  = (mk & (n < firstLoc)) ? n : firstLoc;
  }
  for (int m = 16; m; m >>= 1) {
    const int o = __shfl_xor(firstLoc, m, 32);
    firstLoc = (o < firstLoc) ? o : firstLoc;
  }
  if (lane == 0) swi[wid] = firstLoc;
  __syncthreads();
  if (wid == 0) {
    int v = swi[lane];
    for (int m = 16; m; m >>= 1) {
      const int o = __shfl_xor(v, m, 32);
      v = (o < v) ? o : v;
    }
    if (lane == 0) s_last = (v >= N_) ? 0 : v;
  }
  __syncthreads();

  // ---- serial FPS: M steps, fully LDS-resident ----
  for (int step = 0; step < M_; ++step) {
    if (tid == 0) {
      const int last = s_last;
      fps_idx[b * M_ + step] = last;
      s_px = s_xyz[last * 3 + 0];
      s_py = s_xyz[last * 3 + 1];
      s_pz = s_xyz[last * 3 + 2];
    }
    __syncthreads();
    const float px = s_px, py = s_py, pz = s_pz;

    float bestV = -__builtin_inff();
    int   bestI = 0x7fffffff;
    for (int n = tid; n < N_; n += 1024) {
      const float dx = s_xyz[n * 3 + 0] - px;
      const float dy = s_xyz[n * 3 + 1] - py;
      const float dz = s_xyz[n * 3 + 2] - pz;
      const float dd = dx * dx + dy * dy + dz * dz;
      const float nm = fminf(s_mind[n], dd);
      s_mind[n] = nm;
      // ascending n per thread: strict '>' implements the low-index tie-break
      const bool better = nm > bestV;
      bestV = better ? nm : bestV;
      bestI = better ? n : bestI;
    }
    for (int m = 16; m; m >>= 1) {
      const float ov = __shfl_xor(bestV, m, 32);
      const int   oi = __shfl_xor(bestI, m, 32);
      const bool better = (ov > bestV) | ((ov == bestV) & (oi < bestI));
      bestV = better ? ov : bestV;
      bestI = better ? oi : bestI;
    }
    if (lane == 0) { swv[wid] = bestV; swi[wid] = bestI; }
    __syncthreads();
    if (wid == 0) {
      float v = swv[lane];
      int   i = swi[lane];
      for (int m = 16; m; m >>= 1) {
        const float ov = __shfl_xor(v, m, 32);
        const int   oi = __shfl_xor(i, m, 32);
        const bool better = (ov > v) | ((ov == v) & (oi < i));
        v = better ? ov : v;
        i = better ? oi : i;
      }
      if (lane == 0) s_last = i;
    }
    __syncthreads();
  }
}

// ---------------------------------------------------------------------------
// Kernel 2: three_nn + inverse-distance weights via WMMA f32 16x16x4.
// score[m][n] = -2*a_m . b_n + |b_n|^2  (per-row-constant |a_m|^2 dropped;
// preserves per-row top-3 ordering). One wave owns 16 sampled rows and walks
// 1250 column tiles of 16 cloud points. ISA operand layouts:
//   A 16x4 f32: lanes 0-15 -> (K0,K1), lanes 16-31 -> (K2,K3), row = lane&15
//   B 4x16 f32: lanes 0-15 -> (K0,K1), lanes 16-31 -> (K2,K3), col = lane&15
//   C/D 16x16 f32: VGPR i -> row i (lanes 0-15) / row i+8 (lanes 16-31)
// Top-3 kept branch-free per lane (streaming: strict '<' only), then merged
// across the 16 lanes sharing each row group via shfl_xor 1,2,4,8.
// ---------------------------------------------------------------------------
__global__ __launch_bounds__(256) void three_nn_kernel(
    const float* __restrict__ pc, const int* __restrict__ fps_idx,
    int* __restrict__ idx3, float* __restrict__ w3) {
  const int tid  = threadIdx.x;
  const int lane = tid & 31;
  const int wave = blockIdx.x * (256 >> 5) + (tid >> 5);
  const int tilesPerBatch = M_ / 16;  // 64
  const int b    = wave / tilesPerBatch;
  const int rt   = wave - b * tilesPerBatch;
  const int half = lane >> 4;
  const int l15  = lane & 15;

  const float* pcb = pc + (size_t)b * N_ * 3;

  // A operand (loop invariant): row = rt*16 + l15
  const int mrow = rt * 16 + l15;
  const int aidx = fps_idx[b * M_ + mrow];
  const float* ap = pcb + (size_t)aidx * 3;
  const float ax = ap[0], ay = ap[1], az = ap[2];
  v2f A;
  A.x = half ? (-2.0f * az) : (-2.0f * ax);  // K2 : K0
  A.y = half ? 1.0f         : (-2.0f * ay);  // K3 : K1

  float td[8][3];
  int   ti[8][3];
#pragma unroll
  for (int r = 0; r < 8; ++r)
#pragma unroll
    for (int k = 0; k < 3; ++k) { td[r][k] = __builtin_inff(); ti[r][k] = 0x7fffffff; }

#pragma unroll 2
  for (int t = 0; t < N_ / 16; ++t) {   // 1250 column tiles, exact
    const int n = t * 16 + l15;
    const float* bp = pcb + (size_t)n * 3;
    // prefetch 8 tiles (128 points) ahead, clamped in-range
    const int pn = (n + 128 < N_) ? (n + 128) : (N_ - 1);
    __builtin_prefetch(pcb + (size_t)pn * 3, 0, 1);
    const float bx = bp[0], by = bp[1], bz = bp[2];
    const float bn2 = bx * bx + by * by + bz * bz;
    v8f c = {0.f, 0.f, 0.f, 0.f, 0.f, 0.f, 0.f, 0.f};
#if USE_WMMA
    v2f Bv;
    Bv.x = half ? bz  : bx;   // K2 : K0
    Bv.y = half ? bn2 : by;   // K3 : K1
    c = __builtin_amdgcn_wmma_f32_16x16x4_f32(
        /*neg_a=*/false, A, /*neg_b=*/false, Bv,
        /*c_mod=*/(short)0, c, /*reuse_a=*/false, /*reuse_b=*/false);
#else
#pragma unroll
    for (int r = 0; r < 8; ++r) {
      const int row = half * 8 + r;     // source lane holding that row's xyz
      const float rx = __shfl(ax, row, 32);
      const float ry = __shfl(ay, row, 32);
      const float rz = __shfl(az, row, 32);
      c[r] = -2.0f * (rx * bx + ry * by + rz * bz) + bn2;
    }
#endif
#pragma unroll
    for (int r = 0; r < 8; ++r) ins3_stream(c[r], n, td[r], ti[r]);
  }

  // merge per-row top-3 across the 16 lanes sharing each row group
#pragma unroll
  for (int mskk = 1; mskk <= 8; mskk <<= 1) {
#pragma unroll
    for (int r = 0; r < 8; ++r) {
      float pd[3]; int pi[3];
#pragma unroll
      for (int k = 0; k < 3; ++k) {
        pd[k] = __shfl_xor(td[r][k], mskk, 32);
        pi[k] = __shfl_xor(ti[r][k], mskk, 32);
      }
#pragma unroll
      for (int k = 0; k < 3; ++k) ins3_lex(pd[k], pi[k], td[r], ti[r]);
    }
  }

  // lanes 0 and 16 hold the merged result for rows [0..7] / [8..15]
  if (l15 == 0) {
#pragma unroll
    for (int r = 0; r < 8; ++r) {
      const int row = rt * 16 + half * 8 + r;
      const int gi  = fps_idx[b * M_ + row];
      const float* gp = pcb + (size_t)gi * 3;
      const float gx = gp[0], gy = gp[1], gz = gp[2];
      float wv[3];
      float wsum = 0.f;
#pragma unroll
      for (int k = 0; k < 3; ++k) {
        const int nn = ti[r][k];
        const float* np = pcb + (size_t)nn * 3;
        const float dx = np[0] - gx, dy = np[1] - gy, dz = np[2] - gz;
        const float dd = dx * dx + dy * dy + dz * dz;   // exact d2 (incl. |a|^2)
        const float dist = sqrtf(fmaxf(dd, 0.0f));
        wv[k] = 1.0f / (dist + 1e-8f);
        wsum += wv[k];
        idx3[((size_t)b * M_ + row) * 3 + k] = nn;
      }
#pragma unroll
      for (int k = 0; k < 3; ++k)
        w3[((size_t)b * M_ + row) * 3 + k] = wv[k] / wsum;
    }
  }
}

// ---------------------------------------------------------------------------
// Kernel 3: inverse-distance-weighted feature interpolation, transposed:
// one block per (b,c) so output writes are coalesced and all 3 gathers per m
// land inside this block's single 80 KB feature row (L2/cacheline friendly).
// out[b,c,m] = sum_k w[b,m,k] * feats[b, c, idx3[b,m,k]]
// ---------------------------------------------------------------------------
__global__ __launch_bounds__(256) void interp_kernel(
    const float* __restrict__ feats, const int* __restrict__ idx3,
    const float* __restrict__ w3, float* __restrict__ out) {
  const int bc = blockIdx.x;         // b*C_ + c
  const int b  = bc / C_;
  const int c  = bc - b * C_;
  const float* f = feats + ((size_t)b * C_ + c) * N_;
  float*       o = out   + ((size_t)b * C_ + c) * M_;
  const int*   ib = idx3 + (size_t)b * M_ * 3;
  const float* wb = w3   + (size_t)b * M_ * 3;
  for (int m = threadIdx.x; m < M_; m += 256) {
    const int   n0 = ib[m * 3 + 0], n1 = ib[m * 3 + 1], n2 = ib[m * 3 + 2];
    const float w0 = wb[m * 3 + 0], w1 = wb[m * 3 + 1], w2 = wb[m * 3 + 2];
    o[m] = w0 * f[n0] + w1 * f[n1] + w2 * f[n2];
  }
}

// ---------------------------------------------------------------------------
extern "C" void kernel_launch(void* const* d_in, const int* in_sizes, int n_in,
                              void* d_out, int out_size, void* d_ws, size_t ws_size,
                              hipStream_t stream) {
  const float* pc    = (const float*)d_in[0];  // (B,N,3)
  const float* feats = (const float*)d_in[1];  // (B,C,N)
  const float* obj   = (const float*)d_in[2];  // (B,2,N)
  const float* grasp = (const float*)d_in[3];  // (B,N)
  float* out = (float*)d_out;                  // (B,C,M)

  char* ws = (char*)d_ws;
  size_t off = 0;
  int* fps = (int*)(ws + off);
  off += (size_t)B_ * M_ * 4;        off = (off + 255) & ~(size_t)255;
  int* idx3 = (int*)(ws + off);
  off += (size_t)B_ * M_ * 3 * 4;    off = (off + 255) & ~(size_t)255;
  float* w3 = (float*)(ws + off);
  off += (size_t)B_ * M_ * 3 * 4;    // total ~116 KB

  fps_kernel<<<B_, 1024, 0, stream>>>(pc, obj, grasp, fps);
  three_nn_kernel<<<(B_ * (M_ / 16)) / 8, 256, 0, stream>>>(pc, fps, idx3, w3);
  interp_kernel<<<B_ * C_, 256, 0, stream>>>(feats, idx3, w3, out);
}